// Voxelizer_89618787598659
// MI455X (gfx1250) — compile-verified
//
#include <hip/hip_runtime.h>

typedef __attribute__((ext_vector_type(2))) float v2f;
typedef __attribute__((ext_vector_type(8))) float v8f;

#define GW 64
#define GH 48
#define GD 12
#define NPTS 131072
#define NBATCH 64
#define INV_VOXEL 10.0f

// ORIGIN = (-3.2, -2.4, -0.6)

__global__ void vox_zero_kernel(float* __restrict__ out, int n) {
    int i = blockIdx.x * blockDim.x + threadIdx.x;
    int stride = gridDim.x * blockDim.x;
    for (; i < n; i += stride) out[i] = 0.0f;
}

// Branchless B-matrix tile fetch: column `row` of B = [x, y, z, 1]^T of point
// (tile*16 + row). Layout (mirrors 16x4 A): lanes 0-15 hold K=0 (v0), K=1 (v1);
// lanes 16-31 hold K=2 (v0), K=3 (v1). NT loads: point stream is single-use.
__device__ __forceinline__ v2f load_btile(const float* __restrict__ base,
                                          int tile, int row, int hi) {
    const float* __restrict__ pp = base + (size_t)(tile * 16 + row) * 3;
    // lanes 0-15: pp[0] (=x); lanes 16-31: pp[2] (=z)
    float c0 = __builtin_nontemporal_load(pp + 2 * hi);
    // all lanes load pp[1] (=y, always in bounds), hi lanes select constant 1.0
    float yv = __builtin_nontemporal_load(pp + 1);
    float c1 = hi ? 1.0f : yv;
    v2f bm; bm.x = c0; bm.y = c1;
    return bm;
}

__device__ __forceinline__ void emit_voxel(const v8f& acc, int b, int hi,
                                           float* __restrict__ out) {
    // Lanes 0-15 hold D rows 0-7 for their column => acc[0..2] = (x,y,z) of
    // that lane's point. Lanes 16-31 hold rows 8-15 (zeros) -> masked off.
    if (!hi) {
        const float cx = acc[0];
        const float cy = acc[1];
        const float cz = acc[2];
        const int ix = (int)cx;   // trunc toward zero == jnp.trunc -> int32
        const int iy = (int)cy;
        const int iz = (int)cz;
        const bool valid = (ix >= 0) & (ix < GW) &
                           (iy >= 0) & (iy < GH) &
                           (iz >= 0) & (iz < GD);
        if (valid) {
            const size_t flat =
                (((size_t)b * GW + (size_t)ix) * GH + (size_t)iy) * GD + (size_t)iz;
            out[flat] = 1.0f;   // idempotent scatter-max of 1 on zeroed grid
        }
    }
}

__global__ __launch_bounds__(256) void vox_wmma_kernel(
    const float* __restrict__ points,   // (B, N, 3)
    const float* __restrict__ rpos,     // (B, 3)
    const float* __restrict__ rquat,    // (B, 4) w,x,y,z
    float* __restrict__ out)            // (B, GW, GH, GD) occupancy (float)
{
    const int b    = blockIdx.y;
    const int lane = threadIdx.x & 31;
    const int row  = lane & 15;        // A row / B,D column
    const int hi   = lane >> 4;        // 0: lanes 0-15, 1: lanes 16-31

    const int waveInBlock = threadIdx.x >> 5;
    const int gwave  = blockIdx.x * (blockDim.x >> 5) + waveInBlock;
    const int nwaves = gridDim.x * (blockDim.x >> 5);

    // ---- Per-batch affine transform: coords_f = Mm * p + cc ----
    const float tx = rpos[b * 3 + 0];
    const float ty = rpos[b * 3 + 1];
    const float tz = rpos[b * 3 + 2];
    const float qw =  rquat[b * 4 + 0];     // inverse (conjugate) quaternion
    const float qx = -rquat[b * 4 + 1];
    const float qy = -rquat[b * 4 + 2];
    const float qz = -rquat[b * 4 + 3];

    float R[3][3];
    R[0][0] = 1.0f - 2.0f * (qy * qy + qz * qz);
    R[0][1] = 2.0f * (qx * qy - qw * qz);
    R[0][2] = 2.0f * (qx * qz + qw * qy);
    R[1][0] = 2.0f * (qx * qy + qw * qz);
    R[1][1] = 1.0f - 2.0f * (qx * qx + qz * qz);
    R[1][2] = 2.0f * (qy * qz - qw * qx);
    R[2][0] = 2.0f * (qx * qz - qw * qy);
    R[2][1] = 2.0f * (qy * qz + qw * qx);
    R[2][2] = 1.0f - 2.0f * (qx * qx + qy * qy);

    const float origin[3] = {-3.2f, -2.4f, -0.6f};
    float Mm[3][3];
    float cc[3];
#pragma unroll
    for (int i = 0; i < 3; ++i) {
        Mm[i][0] = R[i][0] * INV_VOXEL;
        Mm[i][1] = R[i][1] * INV_VOXEL;
        Mm[i][2] = R[i][2] * INV_VOXEL;
        cc[i] = (-(R[i][0] * tx + R[i][1] * ty + R[i][2] * tz) - origin[i]) * INV_VOXEL;
    }

    // ---- A matrix (16x4 f32, 2 VGPRs): rows 0-2 = [M row | c], rest zero.
    v2f amat; amat.x = 0.0f; amat.y = 0.0f;
    if (row < 3) {
        if (!hi) { amat.x = Mm[row][0]; amat.y = Mm[row][1]; }
        else     { amat.x = Mm[row][2]; amat.y = cc[row];    }
    }

    const float* __restrict__ base = points + (size_t)b * (size_t)NPTS * 3;
    const int T = NPTS / 16;   // 8192 tiles of 16 points

    // 2-way unrolled: two independent tiles in flight per iteration
    // (2x memory-level parallelism; WMMAs overlap the second load group).
    int t = gwave;
    for (; t + nwaves < T; t += 2 * nwaves) {
        const int t0 = t;
        const int t1 = t + nwaves;

        v2f b0 = load_btile(base, t0, row, hi);
        v2f b1 = load_btile(base, t1, row, hi);

        // Prefetch the first tile (t0) of the next iteration's pair (-> global_prefetch_b8)
        const int tn = t + 2 * nwaves;
        if (tn < T) {
            __builtin_prefetch((const void*)(base + (size_t)(tn * 16 + row) * 3), 0, 1);
        }

        v8f acc0 = {};
        acc0 = __builtin_amdgcn_wmma_f32_16x16x4_f32(
            false, amat, false, b0, (short)0, acc0, false, false);
        v8f acc1 = {};
        acc1 = __builtin_amdgcn_wmma_f32_16x16x4_f32(
            false, amat, false, b1, (short)0, acc1, false, false);

        emit_voxel(acc0, b, hi, out);
        emit_voxel(acc1, b, hi, out);
    }
    // tail (single tile)
    for (; t < T; t += nwaves) {
        v2f b0 = load_btile(base, t, row, hi);
        v8f acc0 = {};
        acc0 = __builtin_amdgcn_wmma_f32_16x16x4_f32(
            false, amat, false, b0, (short)0, acc0, false, false);
        emit_voxel(acc0, b, hi, out);
    }
}

extern "C" void kernel_launch(void* const* d_in, const int* in_sizes, int n_in,
                              void* d_out, int out_size, void* d_ws, size_t ws_size,
                              hipStream_t stream) {
    const float* points = (const float*)d_in[0];   // points_world (B,N,3)
    const float* rpos   = (const float*)d_in[1];   // robot_pos   (B,3)
    const float* rquat  = (const float*)d_in[2];   // robot_quat  (B,4)
    float* out = (float*)d_out;                    // (B,64,48,12)

    // 1) zero the occupancy grid
    {
        int n = out_size;
        int blocks = (n + 255) / 256;
        if (blocks > 4096) blocks = 4096;
        vox_zero_kernel<<<dim3(blocks), dim3(256), 0, stream>>>(out, n);
    }

    // 2) transform + voxelize: 8192 blocks (128 x 64 batches), 8 waves/block,
    //    each wave WMMA-transforms tile pairs of 16 points.
    dim3 grid(128, NBATCH);
    vox_wmma_kernel<<<grid, dim3(256), 0, stream>>>(points, rpos, rquat, out);
}